// EqLayerNodeAttr_88656714925233
// MI455X (gfx1250) — compile-verified
//
#include <hip/hip_runtime.h>

// ---------------- CDNA5 types ----------------
typedef __attribute__((ext_vector_type(2))) float    v2f;
typedef __attribute__((ext_vector_type(8))) float    v8f;
typedef __attribute__((ext_vector_type(4))) unsigned u32x4;
typedef __attribute__((ext_vector_type(8))) int      i32x8;
typedef __attribute__((ext_vector_type(4))) int      i32x4;

__device__ __forceinline__ v8f wmma4(v2f a, v2f b, v8f c) {
  // V_WMMA_F32_16X16X4_F32: (neg_a, A, neg_b, B, c_mod, C, reuse_a, reuse_b)
  return __builtin_amdgcn_wmma_f32_16x16x4_f32(false, a, false, b, (short)0, c,
                                               false, false);
}

#define TM   32          // edges per block
#define KIN  608         // mlp_in
#define HID  256
#define NOUT 192         // DOUT
#define FSTR 610         // sFeat row stride (conflict-free A reads)
#define HSTR 258         // sH row stride
#define KC   16          // K panel depth (double buffered)

#if __has_builtin(__builtin_amdgcn_tensor_load_to_lds) && \
    __has_builtin(__builtin_amdgcn_s_wait_tensorcnt)
#define USE_TDM 1
#else
#define USE_TDM 0
#endif

#if USE_TDM
// 2-D tile DMA: ncols contiguous f32 per row, krows rows, row stride = ncols.
__device__ __forceinline__ void tdm_load_2d(const float* gsrc, unsigned lds_byte,
                                            unsigned ncols, unsigned krows) {
  unsigned long long ga = (unsigned long long)(uintptr_t)gsrc;
  u32x4 g0;
  g0.x = 1u;                                   // count=1, user descriptor
  g0.y = lds_byte;                             // LDS destination (bytes)
  g0.z = (unsigned)ga;                         // global_addr[31:0]
  g0.w = (unsigned)(ga >> 32) | (2u << 30);    // global_addr[56:32] | type=2
  i32x8 g1;
  g1[0] = (int)(2u << 16);                                   // data_size = 4B
  g1[1] = (int)((ncols & 0xFFFFu) << 16);                    // tensor_dim0[15:0]
  g1[2] = (int)((ncols >> 16) | ((krows & 0xFFFFu) << 16));  // tdim0[31:16]|tdim1[15:0]
  g1[3] = (int)((krows >> 16) | ((ncols & 0xFFFFu) << 16));  // tdim1[31:16]|tile_dim0
  g1[4] = (int)(krows & 0xFFFFu);                            // tile_dim1 (tile_dim2=0)
  g1[5] = (int)ncols;                                        // tensor_dim0_stride lo32
  g1[6] = 0;                                                 // stride0 hi | stride1 lo
  g1[7] = 0;                                                 // stride1 hi
  i32x4 gz4 = {0, 0, 0, 0};
  i32x8 gz8 = {0, 0, 0, 0, 0, 0, 0, 0};
  // 6-arg toolchain form: (g0, g1, g2, g3, g4, cpol)
  __builtin_amdgcn_tensor_load_to_lds(g0, g1, gz4, gz4, gz8, 0);
}
#endif

__global__ void zero_f32_kernel(float* __restrict__ p, int n) {
  int i = blockIdx.x * blockDim.x + threadIdx.x;
  if (i < n) p[i] = 0.0f;
}

__global__ __launch_bounds__(256) void eq_edge_mlp_kernel(
    const float* __restrict__ x_scalar,   // (N,64)
    const float* __restrict__ x_rot,      // (N,16,8)
    const float* __restrict__ na_scalar,  // (N,16)
    const float* __restrict__ na_rot,     // (N,8,8)
    const int*   __restrict__ edge_index, // (2,E)
    const float* __restrict__ dist_emb,   // (E,64)
    const float* __restrict__ rot,        // (E,4,2,2)
    const float* __restrict__ W1,         // (608,256)
    const float* __restrict__ b1,         // (256,)
    const float* __restrict__ W2,         // (256,192)
    const float* __restrict__ b2,         // (192,)
    float*       __restrict__ out,        // (N,192) pre-zeroed
    int E) {
  __shared__ __align__(16) float sFeat[TM * FSTR];    // 78,080 B (reused as sOut)
  __shared__ __align__(16) float sH[TM * HSTR];       // 33,024 B
  __shared__ __align__(16) float sW[2 * KC * HID];    // 32,768 B (TDM double buffer)
  __shared__ int sRow[TM];
  __shared__ int sCol[TM];

  const int tid  = threadIdx.x;
  const int e0   = blockIdx.x * TM;
  const int wave = tid >> 5;
  const int lane = tid & 31;
  const int ln   = lane & 15;
  const int hi   = lane >> 4;

#if USE_TDM
  const unsigned sW_base =
      (unsigned)(uintptr_t)(__attribute__((address_space(3))) void*)(void*)sW;
  // prefetch GEMM1 panel 0 while we build features
  if (wave == 0) tdm_load_2d(W1, sW_base, HID, KC);
#endif

  if (tid < TM) {
    int ee = e0 + tid; if (ee >= E) ee = E - 1;
    sRow[tid] = edge_index[ee];       // row = source
    sCol[tid] = edge_index[E + ee];   // col = destination
  }
  __syncthreads();

  // ---------- Stage 1: build per-edge input features (32 x 608) ----------
  for (int idx = tid; idx < TM * KIN; idx += 256) {
    int e = idx / KIN;
    int c = idx - e * KIN;
    int ee = e0 + e; if (ee >= E) ee = E - 1;
    float v;
    if (c >= 544) {
      v = dist_emb[(size_t)ee * 64 + (c - 544)];
    } else {
      int side = (c >= 272) ? 1 : 0;           // 0: x_dst (col), 1: x_src (row)
      int c2   = c - side * 272;
      int node = side ? sRow[e] : sCol[e];
      if (c2 < 64) {
        v = x_scalar[(size_t)node * 64 + c2];
      } else if (c2 < 80) {
        v = na_scalar[(size_t)node * 16 + (c2 - 64)];
      } else {
        int t = c2 - 80;                        // 0..191
        int j = t >> 3, l = (t >> 1) & 3, lo = t & 1;
        float u0, u1;
        if (j < 16) {
          const float* p = &x_rot[(size_t)node * 128 + j * 8 + l * 2];
          u0 = p[0]; u1 = p[1];
        } else {
          const float* p = &na_rot[(size_t)node * 64 + (j - 16) * 8 + l * 2];
          u0 = p[0]; u1 = p[1];
        }
        const float* R = &rot[((size_t)ee * 4 + l) * 4];  // rot[l][a][b]=R[a*2+b]
        v = u0 * R[lo * 2 + 0] + u1 * R[lo * 2 + 1];      // r @ rot_inv
      }
    }
    sFeat[e * FSTR + c] = v;
  }
  __syncthreads();

  // ---------- GEMM1: h(32x256) = relu(feat(32x608) @ W1 + b1) ----------
  const int nt0 = wave * 2;   // wave owns N-tiles {2w,2w+1} x M-tiles {0,1}
  v8f acc00 = {0.f,0.f,0.f,0.f,0.f,0.f,0.f,0.f};
  v8f acc01 = acc00, acc10 = acc00, acc11 = acc00;

  const int NP1 = KIN / KC;   // 38 panels
  for (int p = 0; p < NP1; ++p) {
    const int buf = p & 1;
#if USE_TDM
    if (wave == 0) {
      if (p + 1 < NP1) {
        tdm_load_2d(W1 + (size_t)(p + 1) * KC * HID,
                    sW_base + (unsigned)(((p + 1) & 1) * KC * HID * 4), HID, KC);
        __builtin_amdgcn_s_wait_tensorcnt(1);   // panel p landed, p+1 in flight
      } else {
        __builtin_amdgcn_s_wait_tensorcnt(0);
      }
    }
#else
    for (int idx = tid; idx < KC * HID; idx += 256) {
      int k = idx >> 8, n = idx & 255;
      sW[buf * (KC * HID) + k * HID + n] = W1[(size_t)(p * KC + k) * HID + n];
    }
#endif
    __syncthreads();
    const float* Wp = &sW[buf * (KC * HID)];
#pragma unroll
    for (int kk = 0; kk < KC; kk += 4) {
      const int ka = p * KC + kk + 2 * hi;
      v2f a0 = *(const v2f*)&sFeat[(0 * 16 + ln) * FSTR + ka];
      v2f a1 = *(const v2f*)&sFeat[(1 * 16 + ln) * FSTR + ka];
      const float* pb0 = &Wp[(kk + 2 * hi) * HID + (nt0 + 0) * 16 + ln];
      const float* pb1 = &Wp[(kk + 2 * hi) * HID + (nt0 + 1) * 16 + ln];
      v2f bA = {pb0[0], pb0[HID]};
      v2f bB = {pb1[0], pb1[HID]};
      acc00 = wmma4(a0, bA, acc00);
      acc01 = wmma4(a0, bB, acc01);
      acc10 = wmma4(a1, bA, acc10);
      acc11 = wmma4(a1, bB, acc11);
    }
    __syncthreads();
  }

#if USE_TDM
  // prefetch GEMM2 panel 0 (both buffers' GEMM1 readers are past the barriers)
  if (wave == 0) tdm_load_2d(W2, sW_base, NOUT, KC);
#endif

  { // bias + ReLU, store h tile
    const int n0 = (nt0 + 0) * 16 + ln;
    const int n1 = (nt0 + 1) * 16 + ln;
    const float bb0 = b1[n0], bb1 = b1[n1];
#pragma unroll
    for (int r = 0; r < 8; ++r) {
      const int m = r + hi * 8;
      float v;
      v = acc00[r] + bb0; sH[(0 * 16 + m) * HSTR + n0] = v > 0.f ? v : 0.f;
      v = acc01[r] + bb1; sH[(0 * 16 + m) * HSTR + n1] = v > 0.f ? v : 0.f;
      v = acc10[r] + bb0; sH[(1 * 16 + m) * HSTR + n0] = v > 0.f ? v : 0.f;
      v = acc11[r] + bb1; sH[(1 * 16 + m) * HSTR + n1] = v > 0.f ? v : 0.f;
    }
  }
  __syncthreads();

  // ---------- GEMM2: o(32x192) = h(32x256) @ W2 + b2 ----------
  // 24 tiles (mt 0..1, nt 0..11); wave w owns tile ids {w, w+8, w+16}.
  const int idA = wave, idB = wave + 8, idC = wave + 16;
  const int mtA = idA / 12, ntA = idA % 12;
  const int mtB = idB / 12, ntB = idB % 12;
  const int mtC = idC / 12, ntC = idC % 12;
  v8f dA = {0.f,0.f,0.f,0.f,0.f,0.f,0.f,0.f};
  v8f dB = dA, dC = dA;

  const int NP2 = HID / KC;   // 16 panels
  for (int p = 0; p < NP2; ++p) {
    const int buf = p & 1;
#if USE_TDM
    if (wave == 0) {
      if (p + 1 < NP2) {
        tdm_load_2d(W2 + (size_t)(p + 1) * KC * NOUT,
                    sW_base + (unsigned)(((p + 1) & 1) * KC * HID * 4), NOUT, KC);
        __builtin_amdgcn_s_wait_tensorcnt(1);
      } else {
        __builtin_amdgcn_s_wait_tensorcnt(0);
      }
    }
#else
    for (int idx = tid; idx < KC * NOUT; idx += 256) {
      int k = idx / NOUT, n = idx - k * NOUT;
      sW[buf * (KC * HID) + k * NOUT + n] = W2[(size_t)(p * KC + k) * NOUT + n];
    }
#endif
    __syncthreads();
    const float* Wp = &sW[buf * (KC * HID)];
#pragma unroll
    for (int kk = 0; kk < KC; kk += 4) {
      const int ka = p * KC + kk + 2 * hi;
      v2f aA = *(const v2f*)&sH[(mtA * 16 + ln) * HSTR + ka];
      v2f aB = *(const v2f*)&sH[(mtB * 16 + ln) * HSTR + ka];
      v2f aC = *(const v2f*)&sH[(mtC * 16 + ln) * HSTR + ka];
      const float* pbA = &Wp[(kk + 2 * hi) * NOUT + ntA * 16 + ln];
      const float* pbB = &Wp[(kk + 2 * hi) * NOUT + ntB * 16 + ln];
      const float* pbC = &Wp[(kk + 2 * hi) * NOUT + ntC * 16 + ln];
      v2f bA = {pbA[0], pbA[NOUT]};
      v2f bB = {pbB[0], pbB[NOUT]};
      v2f bC = {pbC[0], pbC[NOUT]};
      dA = wmma4(aA, bA, dA);
      dB = wmma4(aB, bB, dB);
      dC = wmma4(aC, bC, dC);
    }
    __syncthreads();
  }

  // epilogue: + b2, park tile in LDS (reuse sFeat; nobody reads it anymore)
  float* sOut = sFeat;  // 32*192 floats
  {
    const int nA = ntA * 16 + ln, nB = ntB * 16 + ln, nC = ntC * 16 + ln;
    const float bbA = b2[nA], bbB = b2[nB], bbC = b2[nC];
#pragma unroll
    for (int r = 0; r < 8; ++r) {
      const int m = r + hi * 8;
      sOut[(mtA * 16 + m) * NOUT + nA] = dA[r] + bbA;
      sOut[(mtB * 16 + m) * NOUT + nB] = dB[r] + bbB;
      sOut[(mtC * 16 + m) * NOUT + nC] = dC[r] + bbC;
    }
  }
  __syncthreads();

  // ---------- scatter: rotate message, segment-sum into nodes ----------
  for (int idx = tid; idx < TM * 64; idx += 256) {   // scalar cols 0..63
    int e = idx >> 6, n = idx & 63;
    if (e0 + e < E)
      atomicAdd(&out[(size_t)sCol[e] * NOUT + n], sOut[e * NOUT + n]);
  }
  for (int idx = tid; idx < TM * 64; idx += 256) {   // 64 (j,l) rot pairs / edge
    int e = idx >> 6, p = idx & 63;
    if (e0 + e >= E) continue;
    int j = p >> 2, l = p & 3;
    int cb = 64 + j * 8 + l * 2;
    float i0 = sOut[e * NOUT + cb];
    float i1 = sOut[e * NOUT + cb + 1];
    const float* R = &rot[((size_t)(e0 + e) * 4 + l) * 4];
    float o0 = i0 * R[0] + i1 * R[2];   // mr @ rot
    float o1 = i0 * R[1] + i1 * R[3];
    size_t ob = (size_t)sCol[e] * NOUT + cb;
    atomicAdd(&out[ob],     o0);
    atomicAdd(&out[ob + 1], o1);
  }
}

extern "C" void kernel_launch(void* const* d_in, const int* in_sizes, int n_in,
                              void* d_out, int out_size, void* d_ws, size_t ws_size,
                              hipStream_t stream) {
  const float* x_scalar  = (const float*)d_in[0];
  const float* x_rot     = (const float*)d_in[1];
  const float* na_scalar = (const float*)d_in[2];
  const float* na_rot    = (const float*)d_in[3];
  const int*   edge_idx  = (const int*)d_in[4];
  const float* dist_emb  = (const float*)d_in[5];
  const float* rot       = (const float*)d_in[6];
  const float* W1        = (const float*)d_in[7];
  const float* b1        = (const float*)d_in[8];
  const float* W2        = (const float*)d_in[9];
  const float* b2        = (const float*)d_in[10];
  float* out = (float*)d_out;

  const int E = in_sizes[4] / 2;

  // d_out is poisoned by the harness; atomics need zeroed output.
  zero_f32_kernel<<<(out_size + 255) / 256, 256, 0, stream>>>(out, out_size);

  const int nblocks = (E + TM - 1) / TM;
  eq_edge_mlp_kernel<<<nblocks, 256, 0, stream>>>(
      x_scalar, x_rot, na_scalar, na_rot, edge_idx, dist_emb, rot,
      W1, b1, W2, b2, out, E);
}